// ControlGCNConv_3143916060939
// MI455X (gfx1250) — compile-verified
//
#include <hip/hip_runtime.h>

typedef __attribute__((ext_vector_type(2))) float v2f;
typedef __attribute__((ext_vector_type(8))) float v8f;

#define N_NODES 100000
#define N_EDGES 1200000
#define D 64

// out[n][d] = b[d]  (bias pre-fused into the scatter-add target), deg[n] = 0
__global__ __launch_bounds__(256) void gcn_init_kernel(float* __restrict__ out,
                                                       const float* __restrict__ b,
                                                       float* __restrict__ deg) {
    int i = blockIdx.x * 256 + threadIdx.x;
    if (i < N_NODES * D) out[i] = b[i & (D - 1)];
    if (i < N_NODES) deg[i] = 0.0f;
}

// deg[row[e]] += 1
__global__ __launch_bounds__(256) void gcn_degree_kernel(const int* __restrict__ row,
                                                         float* __restrict__ deg) {
    int e = blockIdx.x * 256 + threadIdx.x;
    if (e < N_EDGES) atomicAdd(&deg[row[e]], 1.0f);
}

// deg -> deg_inv in place
__global__ __launch_bounds__(256) void gcn_deginv_kernel(float* __restrict__ deg) {
    int n = blockIdx.x * 256 + threadIdx.x;
    if (n < N_NODES) {
        float d = deg[n];
        deg[n] = (d > 0.0f) ? (1.0f / d) : 0.0f;
    }
}

// xw = x @ W using fp32 WMMA (v_wmma_f32_16x16x4_f32).
// 8 waves per block; each wave computes a 16-row x 64-col tile.
// A (16x4 f32): lanes 0-15 -> M=lane, V0=K0,V1=K1 ; lanes 16-31 -> M=lane-16, V0=K2,V1=K3
// B (4x16 f32): lanes 0-15 -> N=lane, V0=K0,V1=K1 ; lanes 16-31 -> N=lane-16, V0=K2,V1=K3
// C/D (16x16 f32): VGPR v -> M = v + 8*(lane>=16), N = lane%16
//
// W is pre-staged in LDS in per-lane fragment layout: bfrag[(kg*4+nt)*32 + lane]
// is the aligned float2 that lane `lane` feeds to WMMA for k-group kg, n-tile nt.
// Inner loop B access = one ds_load_b64 per fragment, conflict-free, no repacking.
__global__ __launch_bounds__(256) void gcn_gemm_wmma_kernel(const float* __restrict__ x,
                                                            const float* __restrict__ W,
                                                            float* __restrict__ xw) {
    __shared__ v2f bfrag[16 * 4 * 32];   // 16 KB: [kg][nt][lane]

    // Stage W -> fragment layout (permutation of the 64x64 matrix).
    for (int fp = threadIdx.x; fp < 16 * 4 * 32; fp += 256) {
        const int kg = fp >> 7;               // 0..15
        const int nt = (fp >> 5) & 3;         // 0..3
        const int ln = fp & 31;               // fragment lane
        const int kb = (ln >> 4) << 1;        // 0 | 2
        const int nn = (nt << 4) + (ln & 15); // column 0..63
        const int k  = (kg << 2) + kb;        // row of W (K index)
        v2f v;
        v.x = W[k * D + nn];
        v.y = W[(k + 1) * D + nn];
        bfrag[fp] = v;
    }
    __syncthreads();

    const int wave = threadIdx.x >> 5;
    const int lane = threadIdx.x & 31;
    const int tile = blockIdx.x * 8 + wave;   // 16-row tile index
    const int m0 = tile * 16;
    if (m0 >= N_NODES) return;                // wave-uniform: EXEC all-ones inside

    const int mrow = lane & 15;
    const int kb = (lane >> 4) << 1;          // 0 for lanes 0-15, 2 for lanes 16-31

    const float* __restrict__ xrow = x + (size_t)(m0 + mrow) * D;

    v8f c0 = {}, c1 = {}, c2 = {}, c3 = {};

#pragma unroll
    for (int kg = 0; kg < 16; ++kg) {
        // A fragment: aligned 8B load of two consecutive K values of this lane's row
        const v2f a = *reinterpret_cast<const v2f*>(xrow + (kg << 2) + kb);

        // B fragments: direct ds_load_b64 from staged layout
        const int base = (kg << 7) + lane;    // kg*128 + lane
        const v2f b0 = bfrag[base];
        const v2f b1 = bfrag[base + 32];
        const v2f b2 = bfrag[base + 64];
        const v2f b3 = bfrag[base + 96];

        c0 = __builtin_amdgcn_wmma_f32_16x16x4_f32(false, a, false, b0, (short)0, c0, false, false);
        c1 = __builtin_amdgcn_wmma_f32_16x16x4_f32(false, a, false, b1, (short)0, c1, false, false);
        c2 = __builtin_amdgcn_wmma_f32_16x16x4_f32(false, a, false, b2, (short)0, c2, false, false);
        c3 = __builtin_amdgcn_wmma_f32_16x16x4_f32(false, a, false, b3, (short)0, c3, false, false);
    }

    const int rbase = m0 + (lane >> 4) * 8;
#pragma unroll
    for (int v = 0; v < 8; ++v) {
        float* __restrict__ o = xw + (size_t)(rbase + v) * D + mrow;
        o[0]  = c0[v];
        o[16] = c1[v];
        o[32] = c2[v];
        o[48] = c3[v];
    }
}

// 16 lanes per edge: float4 gather of xw[row[e]], scale by deg_inv[row[e]],
// scatter-add into out[col[e]] with global_atomic_add_f32 (L2-resident).
__global__ __launch_bounds__(256) void gcn_scatter_kernel(const int* __restrict__ row,
                                                          const int* __restrict__ col,
                                                          const float* __restrict__ xw,
                                                          const float* __restrict__ deginv,
                                                          float* __restrict__ out) {
    const unsigned tid = blockIdx.x * 256u + threadIdx.x;
    const unsigned e = tid >> 4;
    const unsigned l = tid & 15u;
    if (e >= N_EDGES) return;

    const int r = row[e];
    const int c = col[e];
    const float w = deginv[r];

    const float4 v = reinterpret_cast<const float4*>(xw + (size_t)r * D)[l];
    float* __restrict__ dst = out + (size_t)c * D + l * 4;

    atomicAdd(dst + 0, v.x * w);
    atomicAdd(dst + 1, v.y * w);
    atomicAdd(dst + 2, v.z * w);
    atomicAdd(dst + 3, v.w * w);
}

extern "C" void kernel_launch(void* const* d_in, const int* in_sizes, int n_in,
                              void* d_out, int out_size, void* d_ws, size_t ws_size,
                              hipStream_t stream) {
    (void)in_sizes; (void)n_in; (void)out_size; (void)ws_size;

    const float* x  = (const float*)d_in[0];                 // [N, 64]
    const int*   ei = (const int*)d_in[1];                   // [2, E] (JAX w/o x64 -> int32)
    const float* W  = (const float*)d_in[2];                 // [64, 64]
    const float* b  = (const float*)d_in[3];                 // [64]

    const int* row = ei;               // edge_index[0]
    const int* col = ei + N_EDGES;     // edge_index[1]

    float* out = (float*)d_out;                              // [N, 64]
    float* deg = (float*)d_ws;                               // N floats (becomes deg_inv)
    float* xw  = (float*)((char*)d_ws + (size_t)N_NODES * sizeof(float)); // N*64 floats (16B aligned)

    // 1) out = b (broadcast), deg = 0
    gcn_init_kernel<<<(N_NODES * D + 255) / 256, 256, 0, stream>>>(out, b, deg);
    // 2) degree scatter-sum
    gcn_degree_kernel<<<(N_EDGES + 255) / 256, 256, 0, stream>>>(row, deg);
    // 3) deg -> deg_inv
    gcn_deginv_kernel<<<(N_NODES + 255) / 256, 256, 0, stream>>>(deg);
    // 4) xw = x @ W  (fp32 WMMA; 6250 row-tiles, 8 per block)
    gcn_gemm_wmma_kernel<<<(6250 + 7) / 8, 256, 0, stream>>>(x, W, xw);
    // 5) weighted scatter-add of messages into out
    gcn_scatter_kernel<<<(N_EDGES * 16) / 256, 256, 0, stream>>>(row, col, xw, deg, out);
}